// CPCModel_NCE_13778255085711
// MI455X (gfx1250) — compile-verified
//
#include <hip/hip_runtime.h>
#include <hip/hip_bf16.h>

typedef float v2f __attribute__((ext_vector_type(2)));
typedef float v8f __attribute__((ext_vector_type(8)));

#define WMMA_F32(a, b, c) \
  __builtin_amdgcn_wmma_f32_16x16x4_f32(false, (a), false, (b), (short)0, (c), false, false)

// ---------------- Layer 1: direct conv, cin=1, K=10, S=5, P=2 ----------------
__global__ __launch_bounds__(256) void k_conv1(const float* __restrict__ x,
                                               const float* __restrict__ w,
                                               const float* __restrict__ bias,
                                               float* __restrict__ y) {
  const int T = 20480, L = 4095;
  __shared__ float xs[645];  // 127*5 + 10
  const int b = blockIdx.y;
  const int t0 = blockIdx.x * 128;
  const int base = t0 * 5 - 2;
  for (int i = threadIdx.x; i < 645; i += 256) {
    int gi = base + i;
    xs[i] = (gi >= 0 && gi < T) ? x[(long)b * T + gi] : 0.0f;
  }
  __syncthreads();
  for (int r = 0; r < 256; ++r) {
    int idx = r * 256 + threadIdx.x;  // over 512*128
    int o = idx >> 7;
    int tt = idx & 127;
    int t = t0 + tt;
    if (t >= L) continue;
    float acc = bias[o];
    const float* wo = w + o * 10;
    int s = tt * 5;
#pragma unroll
    for (int j = 0; j < 10; ++j) acc += wo[j] * xs[s + j];
    y[((long)b * 512 + o) * L + t] = fmaxf(acc, 0.0f);
  }
}

// ------------- per-channel sum / sumsq over (B, L), one block per channel -------------
__global__ __launch_bounds__(256) void k_stats(const float* __restrict__ h, int L,
                                               float* __restrict__ osum,
                                               float* __restrict__ osq) {
  const int c = blockIdx.x;
  const int n = 32 * L;
  float s = 0.f, s2 = 0.f;
  for (int idx = threadIdx.x; idx < n; idx += 256) {
    int b = idx / L;
    int t = idx - b * L;
    float v = h[((long)b * 512 + c) * L + t];
    s += v;
    s2 += v * v;
  }
  __shared__ float r1[256], r2[256];
  r1[threadIdx.x] = s;
  r2[threadIdx.x] = s2;
  __syncthreads();
  for (int off = 128; off > 0; off >>= 1) {
    if (threadIdx.x < off) {
      r1[threadIdx.x] += r1[threadIdx.x + off];
      r2[threadIdx.x] += r2[threadIdx.x + off];
    }
    __syncthreads();
  }
  if (threadIdx.x == 0) { osum[c] = r1[0]; osq[c] = r2[0]; }
}

// ------------- BN coefficients: scale = g*rsqrt(var+eps), shift = b - mean*scale -------------
__global__ void k_bncoef(const float* __restrict__ sum, const float* __restrict__ sq,
                         const float* __restrict__ g, const float* __restrict__ bt,
                         float inv_n, float* __restrict__ sc, float* __restrict__ sh) {
  int c = threadIdx.x;
  float mean = sum[c] * inv_n;
  float var = sq[c] * inv_n - mean * mean;
  float s = g[c] * rsqrtf(var + 1e-5f);
  sc[c] = s;
  sh[c] = bt[c] - mean * s;
}

// ------------- WMMA implicit-GEMM conv: cin=cout=512, BN folded into input load -------------
// Block = 8 waves. Each wave: 16 out-channels x 16 output positions; K = 512*KERN in steps of 4.
template <int KERN, int STRIDE, int PAD>
__global__ __launch_bounds__(256) void k_conv_wmma(
    const float* __restrict__ x, const float* __restrict__ insc,
    const float* __restrict__ insh, const float* __restrict__ w,
    const float* __restrict__ bias, float* __restrict__ y, int Lin, int Lout) {
  constexpr int TW = 15 * STRIDE + KERN;  // input span for 16 positions
  constexpr int CHUNK = 8;                // input channels staged per iteration
  constexpr int NLINE = (TW + 31) / 32;   // 128B cachelines per row (approx)
  __shared__ __align__(16) float xs[CHUNK * TW];
  const int b = blockIdx.z;
  const int t0 = blockIdx.x * 16;
  const int lane = threadIdx.x & 31;
  const int wv = threadIdx.x >> 5;
  const int o0 = blockIdx.y * 128 + wv * 16;
  const int n = lane & 15;
  const int koff = (lane < 16) ? 0 : 2;   // A/B f32 layout: hi half holds K = {2,3}
  const int hi8 = (lane < 16) ? 0 : 8;    // C/D: hi half holds rows M+8
  const int base = t0 * STRIDE - PAD;
  v8f acc = {};
  for (int c0 = 0; c0 < 512; c0 += CHUNK) {
    __syncthreads();
    for (int i = threadIdx.x; i < CHUNK * TW; i += 256) {
      int cc = i / TW;
      int ti = i - cc * TW;
      int c = c0 + cc;
      int gi = base + ti;
      float v = 0.f;
      if (gi >= 0 && gi < Lin)
        v = insc[c] * x[((long)b * 512 + c) * Lin + gi] + insh[c];
      xs[i] = v;
    }
    // software prefetch of next chunk's input rows (global_prefetch_b8)
    if (c0 + CHUNK < 512 && threadIdx.x < CHUNK * NLINE) {
      int cc = threadIdx.x / NLINE;
      int jj = threadIdx.x - cc * NLINE;
      int gi = base + jj * 32;
      if (gi >= 0 && gi < Lin)
        __builtin_prefetch(&x[((long)b * 512 + (c0 + CHUNK + cc)) * Lin + gi], 0, 0);
    }
    __syncthreads();
    constexpr int STEPS = CHUNK * KERN / 4;
#pragma unroll
    for (int j = 0; j < STEPS; ++j) {
      int cc = (KERN == 4) ? j : (j >> 1);
      int tb = (KERN == 4) ? 0 : ((j & 1) * 4);
      int kk = tb + koff;
      v2f a = *(const v2f*)(w + ((long)(o0 + n) * 512 + (c0 + cc)) * KERN + kk);
      v2f bm = *(const v2f*)(xs + cc * TW + n * STRIDE + kk);
      acc = WMMA_F32(a, bm, acc);
    }
  }
  int t = t0 + n;
  if (t < Lout) {
#pragma unroll
    for (int r = 0; r < 8; ++r) {
      int o = o0 + r + hi8;
      float v = acc[r] + bias[o];
      y[((long)b * 512 + o) * Lout + t] = fmaxf(v, 0.f);
    }
  }
}

// ------------- z = BN5(h5), transposed to row-major (B*128, 512) -------------
__global__ __launch_bounds__(512) void k_norm_z(const float* __restrict__ h5,
                                                const float* __restrict__ sc,
                                                const float* __restrict__ sh,
                                                float* __restrict__ z) {
  int row = blockIdx.x;  // b*128 + t
  int b = row >> 7, t = row & 127;
  int c = threadIdx.x;
  z[(long)row * 512 + c] = sc[c] * h5[((long)b * 512 + c) * 128 + t] + sh[c];
}

// ------------- quadratic forms: Q[i] = z_i^T W z_i + b·z_i -------------
// Block = 4 waves sharing one 16-row Z tile staged into LDS with
// GLOBAL_LOAD_ASYNC_TO_LDS_B128 (ASYNCcnt). Each wave owns a 128-wide slice
// of the output dimension d (8 d-tiles x 128 K-steps of WMMA f32 16x16x4).
#define QLDSROW 516  // 512 + 4 pad: keeps rows 16B-aligned and banks spread (4*lane mod 64)
__global__ __launch_bounds__(128) void k_qform(const float* __restrict__ Z,
                                               const float* __restrict__ W,
                                               const float* __restrict__ bvec,
                                               float* __restrict__ Qo) {
  __shared__ __align__(16) float zs[16 * QLDSROW];
  __shared__ float redq[4][16];
  const int i0 = blockIdx.x * 16;
  const int tid = threadIdx.x;
  const int lane = tid & 31;
  const int wv = tid >> 5;  // 0..3
  const int nn = lane & 15;
  const int koff = (lane < 16) ? 0 : 2;
  const int hi8 = (lane < 16) ? 0 : 8;

  // Async copy 16 x 512 floats (32KB) of Z into LDS: 2048 B128 transfers,
  // 16 per thread. Tracked by ASYNCcnt; raw copy, no transform needed.
  {
    const unsigned lds_base = (unsigned)(unsigned long long)(void*)&zs[0];
#pragma unroll 4
    for (int it = 0; it < 16; ++it) {
      int e4 = (it * 128 + tid) * 4;  // float index, multiple of 4
      int row = e4 >> 9;
      int col = e4 & 511;
      unsigned lds_addr = lds_base + (unsigned)(row * QLDSROW + col) * 4u;
      int goff = ((i0 + row) * 512 + col) * 4;  // byte offset from Z base
      asm volatile("global_load_async_to_lds_b128 %0, %1, %2"
                   :: "v"(lds_addr), "v"(goff), "s"(Z)
                   : "memory");
    }
    asm volatile("s_wait_asynccnt 0x0" ::: "memory");
  }
  __syncthreads();

  float qp[8] = {0, 0, 0, 0, 0, 0, 0, 0};
  const float* arow = &zs[nn * QLDSROW];  // A operand: row nn of the tile (ds_load_b64)
  for (int dt = 0; dt < 8; ++dt) {
    const int d0 = wv * 128 + dt * 16;
    const int d = d0 + nn;
    const float* wrow = W + (long)d * 512;  // B operand: N=d, K=c (W row-major [d][c])
    const float bd = bvec[d];
    v8f acc = {};
    for (int c0 = 0; c0 < 512; c0 += 4) {
      v2f a = *(const v2f*)(arow + c0 + koff);
      v2f bm = *(const v2f*)(wrow + c0 + koff);
      acc = WMMA_F32(a, bm, acc);
    }
    // fuse: q_i += (Y[i,d] + b[d]) * Z[i,d] over this lane's d
#pragma unroll
    for (int r = 0; r < 8; ++r)
      qp[r] += (acc[r] + bd) * zs[(r + hi8) * QLDSROW + d];
  }
  // intra-wave reduce across the 16 lanes of each half
#pragma unroll
  for (int r = 0; r < 8; ++r) {
    float v = qp[r];
    v += __shfl_xor(v, 1, 16);
    v += __shfl_xor(v, 2, 16);
    v += __shfl_xor(v, 4, 16);
    v += __shfl_xor(v, 8, 16);
    if (nn == 0) redq[wv][r + hi8] = v;  // lane0 -> rows 0..7, lane16 -> rows 8..15
  }
  __syncthreads();
  if (tid < 16)
    Qo[i0 + tid] = redq[0][tid] + redq[1][tid] + redq[2][tid] + redq[3][tid];
}

// ------------- NCE: gather f from Q, log-softmax, logits/labels, per-(k,b) loss -------------
__global__ __launch_bounds__(128) void k_nce(const float* __restrict__ Q,
                                             const int* __restrict__ perm,
                                             float* __restrict__ logits,
                                             float* __restrict__ labels,
                                             float* __restrict__ lpart) {
  const int kk = (int)blockIdx.x / 32 + 1;  // 1..12
  const int b = (int)blockIdx.x & 31;
  const int l = 128 - kk;
  const int t = threadIdx.x;
  const long offk = 352L * ((long)(kk - 1) * 128 - (long)(kk - 1) * kk / 2);
  float rl = 0.f;
  if (t < l) {
    float f[11];
    const float* Qk = Q + (long)(kk - 1) * 4096;
    f[0] = Qk[b * 128 + t + kk];  // positive: z[b, t+k]
    int pt = perm[t + kk];
#pragma unroll
    for (int s = 1; s <= 10; ++s) {
      int bb = (b - s) & 31;  // z_neg_full[b,s-1,t] = z[(b-s)%B, perm[t+k]]
      f[s] = Qk[bb * 128 + pt];
    }
    float m = f[0];
#pragma unroll
    for (int s = 1; s <= 10; ++s) m = fmaxf(m, f[s]);
    float se = 0.f, sf = 0.f;
#pragma unroll
    for (int s = 0; s <= 10; ++s) {
      se += __expf(f[s] - m);
      sf += f[s];
    }
    float lse = m + __logf(se);
    rl = 11.f * lse - sf;  // -sum_s logp over the row
#pragma unroll
    for (int s = 0; s <= 10; ++s) {
      long idx = offk + (long)(b * 11 + s) * l + t;
      logits[idx] = f[s];
      labels[idx] = (s == 0) ? 1.f : 0.f;
    }
  }
  __shared__ float red[128];
  red[t] = rl;
  __syncthreads();
  for (int off = 64; off > 0; off >>= 1) {
    if (t < off) red[t] += red[t + off];
    __syncthreads();
  }
  if (t == 0) lpart[(kk - 1) * 32 + b] = red[0] / ((float)l * 32.f);
}

__global__ void k_loss_final(const float* __restrict__ lpart, float* __restrict__ out) {
  int k = threadIdx.x;  // 0..11
  float s = 0.f;
  for (int b = 0; b < 32; ++b) s += lpart[k * 32 + b];
  out[k] = s;
}

extern "C" void kernel_launch(void* const* d_in, const int* in_sizes, int n_in,
                              void* d_out, int out_size, void* d_ws, size_t ws_size,
                              hipStream_t stream) {
  (void)in_sizes; (void)n_in; (void)out_size; (void)ws_size;
  const float* x = (const float*)d_in[0];
  const float *cw[5], *cb[5], *bg[5], *bb[5];
  for (int i = 0; i < 5; ++i) {
    cw[i] = (const float*)d_in[1 + i];
    cb[i] = (const float*)d_in[6 + i];
    bg[i] = (const float*)d_in[11 + i];
    bb[i] = (const float*)d_in[16 + i];
  }
  const float *lw[12], *lb[12];
  for (int i = 0; i < 12; ++i) {
    lw[i] = (const float*)d_in[21 + i];
    lb[i] = (const float*)d_in[33 + i];
  }
  const int* perm = (const int*)d_in[45];

  float* ws = (float*)d_ws;
  const long A_SZ = 32L * 512 * 4095;  // h1 / h3 / h5
  const long B_SZ = 32L * 512 * 1023;  // h2 / h4
  float* hA = ws;
  float* hB = ws + A_SZ;
  float* z = ws + A_SZ + B_SZ;
  float* Q = z + 4096L * 512;
  float* sum = Q + 12L * 4096;
  float* sq = sum + 512;
  float* sc = sq + 512;
  float* sh = sc + 512;
  float* lpart = sh + 512;

  float* out = (float*)d_out;
  const long NLOG = 513216;  // sum_k 32*11*(128-k)
  float* out_logits = out + 12;
  float* out_labels = out + 12 + NLOG;

  // Layer 1 (direct) -> hA
  k_conv1<<<dim3(32, 32), 256, 0, stream>>>(x, cw[0], cb[0], hA);
  k_stats<<<512, 256, 0, stream>>>(hA, 4095, sum, sq);
  k_bncoef<<<1, 512, 0, stream>>>(sum, sq, bg[0], bb[0], 1.f / (32.f * 4095.f), sc, sh);
  // Layer 2 (K=8,S=4,P=2) -> hB
  k_conv_wmma<8, 4, 2><<<dim3(64, 4, 32), 256, 0, stream>>>(hA, sc, sh, cw[1], cb[1], hB, 4095, 1023);
  k_stats<<<512, 256, 0, stream>>>(hB, 1023, sum, sq);
  k_bncoef<<<1, 512, 0, stream>>>(sum, sq, bg[1], bb[1], 1.f / (32.f * 1023.f), sc, sh);
  // Layer 3 (K=4,S=2,P=2) -> hA
  k_conv_wmma<4, 2, 2><<<dim3(32, 4, 32), 256, 0, stream>>>(hB, sc, sh, cw[2], cb[2], hA, 1023, 512);
  k_stats<<<512, 256, 0, stream>>>(hA, 512, sum, sq);
  k_bncoef<<<1, 512, 0, stream>>>(sum, sq, bg[2], bb[2], 1.f / (32.f * 512.f), sc, sh);
  // Layer 4 (K=4,S=2,P=2) -> hB
  k_conv_wmma<4, 2, 2><<<dim3(17, 4, 32), 256, 0, stream>>>(hA, sc, sh, cw[3], cb[3], hB, 512, 257);
  k_stats<<<512, 256, 0, stream>>>(hB, 257, sum, sq);
  k_bncoef<<<1, 512, 0, stream>>>(sum, sq, bg[3], bb[3], 1.f / (32.f * 257.f), sc, sh);
  // Layer 5 (K=4,S=2,P=1) -> hA
  k_conv_wmma<4, 2, 1><<<dim3(8, 4, 32), 256, 0, stream>>>(hB, sc, sh, cw[4], cb[4], hA, 257, 128);
  k_stats<<<512, 256, 0, stream>>>(hA, 128, sum, sq);
  k_bncoef<<<1, 512, 0, stream>>>(sum, sq, bg[4], bb[4], 1.f / (32.f * 128.f), sc, sh);
  // z (B*128, 512) row-major, BN5 applied
  k_norm_z<<<4096, 512, 0, stream>>>(hA, sc, sh, z);
  // quadratic forms per prediction step (256 tiles of 16 rows each)
  for (int k = 0; k < 12; ++k)
    k_qform<<<256, 128, 0, stream>>>(z, lw[k], lb[k], Q + (long)k * 4096);
  // NCE losses + logits/labels
  k_nce<<<384, 128, 0, stream>>>(Q, perm, out_logits, out_labels, lpart);
  k_loss_final<<<1, 12, 0, stream>>>(lpart, out);
}